// E_GCL_RM_Edge_35304631173700
// MI455X (gfx1250) — compile-verified
//
#include <hip/hip_runtime.h>
#include <hip/hip_bf16.h>

// Problem constants (from reference)
#define NN   8192
#define KK   30
#define DD   320
#define EPSF 1e-8f

typedef __attribute__((ext_vector_type(2))) float v2f;
typedef __attribute__((ext_vector_type(8))) float v8f;

#define NTPW 5              // n-tiles per wave
#define NSTRIDE (4 * 16)    // column stride between a wave's n-tiles (= 64)
#define MROW2 (16 * DD)     // element offset of second M tile (folds into imm)

// ---------------------------------------------------------------------------
// Kernel 0: transpose a [rows x cols] weight matrix into [cols x rows].
// ---------------------------------------------------------------------------
__global__ void transpose_w(const float* __restrict__ W, float* __restrict__ WT,
                            int rows, int cols)
{
    const int idx = blockIdx.x * blockDim.x + threadIdx.x;
    if (idx >= rows * cols) return;
    const int k = idx / cols;       // row of W
    const int n = idx % cols;       // col of W  (coalesced read)
    WT[(size_t)n * rows + k] = W[idx];
}

// ---------------------------------------------------------------------------
// Kernel 1: per-node edge model. One wave32 per node.
// ---------------------------------------------------------------------------
__global__ void edge_kernel(const float* __restrict__ h,
                            const float* __restrict__ coord,
                            const int*   __restrict__ col_idx,   // [N*K]
                            float* __restrict__ agg,             // [N*D] ws
                            float* __restrict__ coord_out)       // [N*3] in d_out
{
    const int wid  = (blockIdx.x * blockDim.x + threadIdx.x) >> 5;
    const int lane = threadIdx.x & 31;
    if (wid >= NN) return;

    const int* mycol = col_idx + (size_t)wid * KK;

    const float cix = coord[wid * 3 + 0];
    const float ciy = coord[wid * 3 + 1];
    const float ciz = coord[wid * 3 + 2];

    // ---- per-neighbor radial / distance score (lanes 0..29) ----
    int   cj = 0;
    float dx = 0.f, dy = 0.f, dz = 0.f;
    float s  = -1e30f;
    if (lane < KK) {
        cj = mycol[lane];
        dx = cix - coord[cj * 3 + 0];
        dy = ciy - coord[cj * 3 + 1];
        dz = ciz - coord[cj * 3 + 2];
        const float radial = dx * dx + dy * dy + dz * dz;
        s = 1.0f / (radial + EPSF);          // DIST_W * dist_score, DIST_W = 1
    }

    // ---- 30 dot products, each lane covers 10 dims of D=320 ----
    float hi[10];
#pragma unroll
    for (int t = 0; t < 10; ++t) hi[t] = h[(size_t)wid * DD + lane + 32 * t];

    for (int j = 0; j < KK; ++j) {
        const int cc = __shfl(cj, j, 32);
        const float* hc = h + (size_t)cc * DD;
        float p = 0.f;
#pragma unroll
        for (int t = 0; t < 10; ++t) p += hi[t] * hc[lane + 32 * t];
#pragma unroll
        for (int off = 16; off > 0; off >>= 1) p += __shfl_xor(p, off, 32);
        if (lane == j) s += p;
    }

    // ---- softmax over lanes 0..29 ----
    float m = s;
#pragma unroll
    for (int off = 16; off > 0; off >>= 1) m = fmaxf(m, __shfl_xor(m, off, 32));
    float e = (lane < KK) ? __expf(s - m) : 0.f;
    float sum = e;
#pragma unroll
    for (int off = 16; off > 0; off >>= 1) sum += __shfl_xor(sum, off, 32);
    const float w = e / sum;

    // ---- coord update: mean over K of w_j * diff_j ----
    float tx = dx * w, ty = dy * w, tz = dz * w;
#pragma unroll
    for (int off = 16; off > 0; off >>= 1) {
        tx += __shfl_xor(tx, off, 32);
        ty += __shfl_xor(ty, off, 32);
        tz += __shfl_xor(tz, off, 32);
    }
    if (lane == 0) {
        const float inv_k = 1.0f / (float)KK;
        coord_out[wid * 3 + 0] = cix + tx * inv_k;
        coord_out[wid * 3 + 1] = ciy + ty * inv_k;
        coord_out[wid * 3 + 2] = ciz + tz * inv_k;
    }

    // ---- aggregate sum_j w_j * h[col_j] ----
    float acc[10];
#pragma unroll
    for (int t = 0; t < 10; ++t) acc[t] = 0.f;
    for (int j = 0; j < KK; ++j) {
        const int   cc = __shfl(cj, j, 32);
        const float wj = __shfl(w,  j, 32);
        const float* hc = h + (size_t)cc * DD;
#pragma unroll
        for (int t = 0; t < 10; ++t) acc[t] += wj * hc[lane + 32 * t];
    }
#pragma unroll
    for (int t = 0; t < 10; ++t) agg[(size_t)wid * DD + lane + 32 * t] = acc[t];
}

// ---------------------------------------------------------------------------
// WMMA tiling: one wave owns TWO 16-row M tiles and FIVE 16-col N tiles.
// Software-pipelined one K-step deep. All lane-dependent address math is
// folded into base pointers at setup; loop indices are uniform and fold into
// the load instructions' immediate offsets (no per-iteration address VALU).
// __launch_bounds__(128,1) gives the allocator the full VGPR file so the
// ~140-register live set (80 acc + 2x14 fragments + pointers) never spills.
// ---------------------------------------------------------------------------
struct Frag {
    v2f a0, a1;
    v2f b[NTPW];
};

__device__ __forceinline__
Frag load_frag(const float* p0, const float* const wr[NTPW], int k)
{
    Frag f;
    f.a0 = *(const v2f*)(p0 + k);
    f.a1 = *(const v2f*)(p0 + MROW2 + k);   // second M tile via imm offset
#pragma unroll
    for (int t = 0; t < NTPW; ++t) f.b[t] = *(const v2f*)(wr[t] + k);
    return f;
}

__device__ __forceinline__
void wmma10(const Frag& f, v8f acc0[NTPW], v8f acc1[NTPW])
{
#pragma unroll
    for (int t = 0; t < NTPW; ++t) {
        acc0[t] = __builtin_amdgcn_wmma_f32_16x16x4_f32(false, f.a0, false, f.b[t],
                                                        (short)0, acc0[t],
                                                        false, false);
        acc1[t] = __builtin_amdgcn_wmma_f32_16x16x4_f32(false, f.a1, false, f.b[t],
                                                        (short)0, acc1[t],
                                                        false, false);
    }
}

// Kernel 2: Y = silu([h | agg] @ W1 + b1)
// W1T is the transposed W1: [D rows][2D cols], W1T[n][k] = W1[k][n]
__global__ void __launch_bounds__(128, 1)
gemm1_silu(const float* __restrict__ h,
           const float* __restrict__ agg,
           const float* __restrict__ W1T,   // [D, 2D]
           const float* __restrict__ b1,    // [D]
           float* __restrict__ Y)           // [N, D] ws
{
    const int wave = threadIdx.x >> 5;
    const int lane = threadIdx.x & 31;
    const int m0   = blockIdx.x * 32;               // 256 blocks, 32 rows each
    const int n0   = wave * 16;                     // first of 5 n-tiles
    const int mr   = lane & 15;
    const int koff = (lane >> 4) << 1;              // 0 or 2

    // fold all lane-dependent offsets into the base pointers
    const float* zh = h   + (size_t)(m0 + mr) * DD + koff;
    const float* za = agg + (size_t)(m0 + mr) * DD + koff;

    const float* wr[NTPW];
#pragma unroll
    for (int t = 0; t < NTPW; ++t)
        wr[t] = W1T + (size_t)(n0 + t * NSTRIDE + mr) * (2 * DD) + koff;

    v8f acc0[NTPW], acc1[NTPW];
#pragma unroll
    for (int t = 0; t < NTPW; ++t) { acc0[t] = (v8f){}; acc1[t] = (v8f){}; }

    // ---- pipelined K loop: half 1 (h, W1T cols 0..319) ----
    Frag cur = load_frag(zh, wr, 0);
    for (int kk = 0; kk < DD - 4; kk += 4) {
        Frag nxt = load_frag(zh, wr, kk + 4);
        __builtin_prefetch(wr[0] + kk + 256, 0, 1);
        wmma10(cur, acc0, acc1);
        cur = nxt;
    }
    {   // bridge into half 2 without draining the pipeline
        Frag nxt = load_frag(za, wr, DD);    // note: A from agg, B at col DD
        // fix A side: load_frag uses same k for A and B; adjust A manually
        nxt.a0 = *(const v2f*)(za + 0);
        nxt.a1 = *(const v2f*)(za + MROW2 + 0);
        wmma10(cur, acc0, acc1);
        cur = nxt;
    }
    // ---- half 2 (agg, W1T cols 320..639) ----
    for (int kk = 0; kk < DD - 4; kk += 4) {
        Frag nxt;
        nxt.a0 = *(const v2f*)(za + kk + 4);
        nxt.a1 = *(const v2f*)(za + MROW2 + kk + 4);
#pragma unroll
        for (int t = 0; t < NTPW; ++t)
            nxt.b[t] = *(const v2f*)(wr[t] + DD + kk + 4);
        __builtin_prefetch(wr[0] + DD + kk + 256, 0, 1);
        wmma10(cur, acc0, acc1);
        cur = nxt;
    }
    wmma10(cur, acc0, acc1);   // epilogue step

#pragma unroll
    for (int t = 0; t < NTPW; ++t) {
        const int nc = n0 + t * NSTRIDE + (lane & 15);
        const float bias = b1[nc];
#pragma unroll
        for (int r = 0; r < 8; ++r) {
            const int mrow = r + ((lane >> 4) << 3);
            float x0 = acc0[t][r] + bias;
            float x1 = acc1[t][r] + bias;
            x0 = x0 / (1.0f + __expf(-x0));         // SiLU
            x1 = x1 / (1.0f + __expf(-x1));
            Y[(size_t)(m0 + mrow)      * DD + nc] = x0;
            Y[(size_t)(m0 + 16 + mrow) * DD + nc] = x1;
        }
    }
}

// Kernel 3: h_out = h + (Y @ W2 + b2)
// W2T is the transposed W2: [D rows][D cols], W2T[n][k] = W2[k][n]
__global__ void __launch_bounds__(128, 1)
gemm2_residual(const float* __restrict__ Y,
               const float* __restrict__ W2T,   // [D, D]
               const float* __restrict__ b2,
               const float* __restrict__ h,
               float* __restrict__ h_out)       // [N, D] in d_out
{
    const int wave = threadIdx.x >> 5;
    const int lane = threadIdx.x & 31;
    const int m0   = blockIdx.x * 32;
    const int n0   = wave * 16;
    const int mr   = lane & 15;
    const int koff = (lane >> 4) << 1;

    const float* yr = Y + (size_t)(m0 + mr) * DD + koff;

    const float* wr[NTPW];
#pragma unroll
    for (int t = 0; t < NTPW; ++t)
        wr[t] = W2T + (size_t)(n0 + t * NSTRIDE + mr) * DD + koff;

    v8f acc0[NTPW], acc1[NTPW];
#pragma unroll
    for (int t = 0; t < NTPW; ++t) { acc0[t] = (v8f){}; acc1[t] = (v8f){}; }

    Frag cur = load_frag(yr, wr, 0);
    for (int kk = 0; kk < DD - 4; kk += 4) {
        Frag nxt = load_frag(yr, wr, kk + 4);
        __builtin_prefetch(wr[0] + kk + 256, 0, 1);
        wmma10(cur, acc0, acc1);
        cur = nxt;
    }
    wmma10(cur, acc0, acc1);   // epilogue step

#pragma unroll
    for (int t = 0; t < NTPW; ++t) {
        const int nc = n0 + t * NSTRIDE + (lane & 15);
        const float bias = b2[nc];
#pragma unroll
        for (int r = 0; r < 8; ++r) {
            const int mrow = r + ((lane >> 4) << 3);
            const size_t i0 = (size_t)(m0 + mrow)      * DD + nc;
            const size_t i1 = (size_t)(m0 + 16 + mrow) * DD + nc;
            h_out[i0] = h[i0] + acc0[t][r] + bias;   // residual
            h_out[i1] = h[i1] + acc1[t][r] + bias;
        }
    }
}

// ---------------------------------------------------------------------------
extern "C" void kernel_launch(void* const* d_in, const int* in_sizes, int n_in,
                              void* d_out, int out_size, void* d_ws, size_t ws_size,
                              hipStream_t stream) {
    const float* h     = (const float*)d_in[0];          // [N, D]
    const float* coord = (const float*)d_in[1];          // [N, 3]
    const float* W1    = (const float*)d_in[2];          // [2D, D]
    const float* b1    = (const float*)d_in[3];          // [D]
    const float* W2    = (const float*)d_in[4];          // [D, D]
    const float* b2    = (const float*)d_in[5];          // [D]
    const int*   eidx  = (const int*)d_in[6];            // [2, N*K]
    const int*   col   = eidx + (size_t)NN * KK;         // second row = col

    float* h_out     = (float*)d_out;                    // [N, D]
    float* coord_out = (float*)d_out + (size_t)NN * DD;  // [N, 3]

    float* agg  = (float*)d_ws;                          // [N, D]
    float* Ybuf = agg  + (size_t)NN * DD;                // [N, D]
    float* W1T  = Ybuf + (size_t)NN * DD;                // [D, 2D]
    float* W2T  = W1T  + (size_t)2 * DD * DD;            // [D, D]

    // Kernel 0: one-shot weight transposes (~1.2 MB total)
    {
        dim3 block(256);
        transpose_w<<<dim3((2 * DD * DD + 255) / 256), block, 0, stream>>>(
            W1, W1T, 2 * DD, DD);
        transpose_w<<<dim3((DD * DD + 255) / 256), block, 0, stream>>>(
            W2, W2T, DD, DD);
    }
    // Kernel 1: one wave per node; 8 waves (256 threads) per block
    {
        dim3 block(256);
        dim3 grid((NN * 32 + 255) / 256);                // 1024 blocks
        edge_kernel<<<grid, block, 0, stream>>>(h, coord, col, agg, coord_out);
    }
    // Kernel 2: 256 blocks x 4 waves; each wave: 2 m-tiles x 5 n-tiles
    {
        dim3 block(128);
        dim3 grid(NN / 32);                              // 256 blocks
        gemm1_silu<<<grid, block, 0, stream>>>(h, agg, W1T, b1, Ybuf);
    }
    // Kernel 3
    {
        dim3 block(128);
        dim3 grid(NN / 32);                              // 256 blocks
        gemm2_residual<<<grid, block, 0, stream>>>(Ybuf, W2T, b2, h, h_out);
    }
}